// OctreeTrilinear_24146306138645
// MI455X (gfx1250) — compile-verified
//
#include <hip/hip_runtime.h>

#define WAVES_PER_BLOCK 8

// Low 32 bits of a generic pointer to __shared__ == LDS byte offset
// (flat LDS aperture only occupies ADDR[63:32]; DS unit uses addr[31:0]).
__device__ __forceinline__ unsigned lds_off_u32(const void* p) {
  return (unsigned)(size_t)p;
}

// ---------------------------------------------------------------------------
// Transpose data (C=32, H) channel-major -> dataT (H, 32) node-major using the
// CDNA5 async LDS data path: global -> LDS -> global with no VGPR round-trip.
// Each wave owns a 32h x 32c tile; ASYNCcnt fences LDS write->read in-wave.
// Stride-33 LDS padding keeps both phases bank-conflict-free.
// ---------------------------------------------------------------------------
__global__ __launch_bounds__(256) void octree_tri_transpose32_async(
    const float* __restrict__ src, float* __restrict__ dst, int H)
{
  __shared__ float t[WAVES_PER_BLOCK][32 * 33];
  const int wave = threadIdx.x >> 5;
  const int lane = threadIdx.x & 31;
  const long long tile = (long long)blockIdx.x * WAVES_PER_BLOCK + wave;
  const int h0 = (int)(tile * 32);
  const int h  = h0 + lane;

  // Coalesced async loads: lane's dword of channel row c -> transposed LDS slot.
  if (h < H) {
#pragma unroll
    for (int c = 0; c < 32; ++c) {
      const float* g = src + (long long)c * H + h;
      const unsigned l = lds_off_u32(&t[wave][c * 33 + lane]);
      asm volatile("global_load_async_to_lds_b32 %0, %1, off"
                   :: "v"(l), "v"(g) : "memory");
    }
  }
  asm volatile("s_wait_asynccnt 0x0" ::: "memory");

  // Coalesced async stores: node row hh (32 channels across lanes) -> dst.
#pragma unroll
  for (int hl = 0; hl < 32; ++hl) {
    const int hh = h0 + hl;
    if (hh < H) {
      const float* g = dst + (long long)hh * 32 + lane;
      const unsigned l = lds_off_u32(&t[wave][lane * 33 + hl]);
      asm volatile("global_store_async_from_lds_b32 %0, %1, off"
                   :: "v"(g), "v"(l) : "memory");
    }
  }
  asm volatile("s_wait_asynccnt 0x0" ::: "memory");
}

// ---------------------------------------------------------------------------
// Main kernel (requires transposed feat (H,32)): one wave32 per 32-point tile.
//   Phase 1 (lane = point):  8 interleaved binary searches + weights -> LDS
//   Phase 2 (8 lanes/point): B128 channel-quad gathers + float4 fmac
//   Phase 3 (lane = point):  LDS transpose -> coalesced (C, N) stores
// ---------------------------------------------------------------------------
__global__ __launch_bounds__(256) void octree_tri_main(
    const float* __restrict__ feat,
    const float* __restrict__ pts, const long long* __restrict__ keys,
    float* __restrict__ out, int N, int H, int nIter)
{
  __shared__ int   sIdx[WAVES_PER_BLOCK][32][8];
  __shared__ float sW  [WAVES_PER_BLOCK][32][8];
  __shared__ __align__(16) float sAcc[WAVES_PER_BLOCK][32 * 36];

  const int wave = threadIdx.x >> 5;
  const int lane = threadIdx.x & 31;
  const long long tile = (long long)blockIdx.x * WAVES_PER_BLOCK + wave;
  const int p0 = (int)(tile * 32);
  const int p  = p0 + lane;

  // ------------------------- Phase 1: lane = point -------------------------
  if (p < N) {
    const float4 q = reinterpret_cast<const float4*>(pts)[p];
    const float xf = q.x - 0.5f;
    const float yf = q.y - 0.5f;
    const float zf = q.z - 0.5f;
    const float xw = floorf(xf), yw = floorf(yf), zw = floorf(zf);
    const float fx = xf - xw, fy = yf - yw, fz = zf - zw;
    const long long basek = (long long)xw + ((long long)yw << 16) +
                            ((long long)zw << 32);

    long long k8[8];
    float     w8[8];
    int       lo[8], hi[8];
#pragma unroll
    for (int k = 0; k < 8; ++k) {
      const int cx = (k >> 2) & 1, cy = (k >> 1) & 1, cz = k & 1;
      k8[k] = basek + (long long)cx + ((long long)cy << 16) + ((long long)cz << 32);
      w8[k] = (cx ? fx : 1.0f - fx) * (cy ? fy : 1.0f - fy) * (cz ? fz : 1.0f - fz);
      lo[k] = 0;
      hi[k] = H;
    }
    // 8 independent search chains interleaved -> 8 loads in flight per lane.
    for (int it = 0; it < nIter; ++it) {
#pragma unroll
      for (int k = 0; k < 8; ++k) {
        if (lo[k] < hi[k]) {
          const int mid = (lo[k] + hi[k]) >> 1;
          const bool cl = keys[mid] < k8[k];
          lo[k] = cl ? (mid + 1) : lo[k];
          hi[k] = cl ? hi[k] : mid;
        }
      }
    }
    float norm = 0.0f;
#pragma unroll
    for (int k = 0; k < 8; ++k) {
      const int pc = (lo[k] < H - 1) ? lo[k] : (H - 1);
      const bool found = (keys[pc] == k8[k]);
      const int id = found ? pc : 0;
      const float wv = found ? w8[k] : 0.0f;
      norm += wv;
      sIdx[wave][lane][k] = id;
      w8[k] = wv;
    }
    const float inv = 1.0f / (norm + 1e-10f);
#pragma unroll
    for (int k = 0; k < 8; ++k) sW[wave][lane][k] = w8[k] * inv;
  } else {
#pragma unroll
    for (int k = 0; k < 8; ++k) {
      sIdx[wave][lane][k] = 0;
      sW  [wave][lane][k] = 0.0f;
    }
  }
  __syncthreads();

  // -------------- Phase 2: 8 lanes per point, B128 row gathers -------------
  const int sub = lane >> 3;   // which of 4 points in this group
  const int cq  = lane & 7;    // channel quad: channels 4*cq .. 4*cq+3
#pragma unroll 2
  for (int g = 0; g < 8; ++g) {
    const int pp = g * 4 + sub;
    float4 acc = make_float4(0.0f, 0.0f, 0.0f, 0.0f);
#pragma unroll
    for (int k = 0; k < 8; ++k) {
      const int   id = sIdx[wave][pp][k];
      const float wv = sW  [wave][pp][k];
      const float4 r = *reinterpret_cast<const float4*>(
          feat + ((long long)id << 5) + (cq << 2));
      acc.x = fmaf(wv, r.x, acc.x);
      acc.y = fmaf(wv, r.y, acc.y);
      acc.z = fmaf(wv, r.z, acc.z);
      acc.w = fmaf(wv, r.w, acc.w);
    }
    *reinterpret_cast<float4*>(&sAcc[wave][pp * 36 + (cq << 2)]) = acc;
  }
  __syncthreads();

  // ------------------------- Phase 3: lane = point -------------------------
  if (p < N) {
#pragma unroll
    for (int c = 0; c < 32; ++c)
      out[(long long)c * N + p] = sAcc[wave][lane * 36 + c];
  }
}

// ---------------------------------------------------------------------------
// Generic fallback (any C, or workspace too small for the transposed table).
// ---------------------------------------------------------------------------
__global__ __launch_bounds__(256) void octree_tri_generic(
    const float* __restrict__ data, const float* __restrict__ pts,
    const long long* __restrict__ keys, float* __restrict__ out,
    int N, int H, int C, int nIter)
{
  const int p = blockIdx.x * blockDim.x + threadIdx.x;
  if (p >= N) return;
  const float4 q = reinterpret_cast<const float4*>(pts)[p];
  const float xf = q.x - 0.5f, yf = q.y - 0.5f, zf = q.z - 0.5f;
  const float xw = floorf(xf), yw = floorf(yf), zw = floorf(zf);
  const float fx = xf - xw, fy = yf - yw, fz = zf - zw;
  const long long basek = (long long)xw + ((long long)yw << 16) +
                          ((long long)zw << 32);

  long long k8[8]; float w8[8]; int lo[8], hi[8];
#pragma unroll
  for (int k = 0; k < 8; ++k) {
    const int cx = (k >> 2) & 1, cy = (k >> 1) & 1, cz = k & 1;
    k8[k] = basek + (long long)cx + ((long long)cy << 16) + ((long long)cz << 32);
    w8[k] = (cx ? fx : 1.0f - fx) * (cy ? fy : 1.0f - fy) * (cz ? fz : 1.0f - fz);
    lo[k] = 0; hi[k] = H;
  }
  for (int it = 0; it < nIter; ++it) {
#pragma unroll
    for (int k = 0; k < 8; ++k) {
      if (lo[k] < hi[k]) {
        const int mid = (lo[k] + hi[k]) >> 1;
        const bool cl = keys[mid] < k8[k];
        lo[k] = cl ? (mid + 1) : lo[k];
        hi[k] = cl ? hi[k] : mid;
      }
    }
  }
  int id8[8]; float norm = 0.0f;
#pragma unroll
  for (int k = 0; k < 8; ++k) {
    const int pc = (lo[k] < H - 1) ? lo[k] : (H - 1);
    const bool found = (keys[pc] == k8[k]);
    id8[k] = found ? pc : 0;
    w8[k]  = found ? w8[k] : 0.0f;
    norm  += w8[k];
  }
  const float inv = 1.0f / (norm + 1e-10f);
#pragma unroll
  for (int k = 0; k < 8; ++k) w8[k] *= inv;
  for (int c = 0; c < C; ++c) {
    float acc = 0.0f;
#pragma unroll
    for (int k = 0; k < 8; ++k)
      acc = fmaf(w8[k], data[(long long)c * H + id8[k]], acc);
    out[(long long)c * N + p] = acc;
  }
}

// ---------------------------------------------------------------------------
extern "C" void kernel_launch(void* const* d_in, const int* in_sizes, int n_in,
                              void* d_out, int out_size, void* d_ws, size_t ws_size,
                              hipStream_t stream)
{
  const float*     data = (const float*)d_in[0];      // (1, C, H, 1) f32
  const float*     pts  = (const float*)d_in[1];      // (N, 4) f32
  const long long* keys = (const long long*)d_in[2];  // (H,) sorted int64
  // d_in[3] = depth: unused by the math (points are pre-transformed).

  const int H = in_sizes[2];
  const int N = in_sizes[1] / 4;
  const int C = (H > 0) ? (in_sizes[0] / H) : 0;
  float* out = (float*)d_out;
  if (N <= 0 || H <= 0 || C <= 0) return;

  // Fixed search trip count: smallest n with 2^n > H keeps all 8 interleaved
  // chains converged (guarded updates make extra iterations no-ops).
  int nIter = 0;
  while ((1LL << nIter) <= (long long)H) ++nIter;

  const size_t needT = (size_t)in_sizes[0] * sizeof(float);
  const bool useT = (C == 32) && (d_ws != nullptr) && (ws_size >= needT);

  if (useT) {
    float* dataT = (float*)d_ws;
    const int tilesT  = (H + 31) / 32;
    const int blocksT = (tilesT + WAVES_PER_BLOCK - 1) / WAVES_PER_BLOCK;
    hipLaunchKernelGGL(octree_tri_transpose32_async, dim3(blocksT), dim3(256),
                       0, stream, data, dataT, H);

    const int tiles  = (N + 31) / 32;
    const int blocks = (tiles + WAVES_PER_BLOCK - 1) / WAVES_PER_BLOCK;
    hipLaunchKernelGGL(octree_tri_main, dim3(blocks), dim3(256), 0, stream,
                       dataT, pts, keys, out, N, H, nIter);
  } else {
    const int blocks = (N + 255) / 256;
    hipLaunchKernelGGL(octree_tri_generic, dim3(blocks), dim3(256), 0, stream,
                       data, pts, keys, out, N, H, C, nIter);
  }
}